// MinErrorInitializer_37220186587513
// MI455X (gfx1250) — compile-verified
//
#include <hip/hip_runtime.h>

// MinErrorInitializer (LSQ step search) for MI455X / gfx1250, wave32.
// One block per row; row staged global->LDS with global_load_async_to_lds_b128
// (ASYNCcnt path); error reduction offloaded to the matrix pipe via
// V_WMMA_F32_16X16X4_F32 with B == ones (D[m,n] += sum_k A[m,k]).
// Argmin carried as a branchless u64 key: (err_bits<<32)|idx, umin-reduced
// (err >= 0 so fp32 bits are order-preserving; ties pick the lower index,
// matching jnp.argmin first-occurrence semantics).

#define N_ITERS   300
#define STEP_MIN  1e-7f
#define C_DIM     1024
#define BLOCK     256
#define NWAVE     (BLOCK / 32)
#define N_TILES   ((N_ITERS + 15) / 16)   // 19 tiles of 16 steps

typedef __attribute__((ext_vector_type(2))) float v2f;
typedef __attribute__((ext_vector_type(8))) float v8f;

__device__ __forceinline__ float sq_err(float xv, float step, float rstep,
                                        float ncf, float pcf) {
    float q = rintf(xv * rstep);          // v_rndne_f32: round-half-even == jnp.round
    q = fminf(fmaxf(q, ncf), pcf);        // clamp AFTER rounding (matches reference)
    float d = fmaf(-q, step, xv);         // x - q*step
    return d * d;                         // P == 2
}

__global__ __launch_bounds__(BLOCK)
void lsq_init_kernel(const float* __restrict__ x,
                     const int* __restrict__ ncp,
                     const int* __restrict__ pcp,
                     float* __restrict__ out) {
    __shared__ float              lds_x[C_DIM];
    __shared__ float              s_max[NWAVE];
    __shared__ unsigned long long s_key[NWAVE];

    const int row  = blockIdx.x;
    const int tid  = threadIdx.x;
    const int lane = tid & 31;
    const int wid  = tid >> 5;

    const float ncf = (float)(*ncp);      // -8
    const float pcf = (float)(*pcp);      //  7

    // ---- Async-stage row into LDS: 16 B per lane, 256 lanes == 4 KB row ----
    {
        unsigned int lds_off =
            (unsigned int)(unsigned long long)(void*)lds_x + (unsigned int)tid * 16u;
        unsigned long long gaddr =
            (unsigned long long)(x + (size_t)row * C_DIM) + (unsigned long long)tid * 16ull;
        asm volatile("global_load_async_to_lds_b128 %0, %1, off"
                     :: "v"(lds_off), "v"(gaddr) : "memory");
        asm volatile("s_wait_asynccnt 0x0" ::: "memory");
    }
    __syncthreads();

    // ---- Row max|x| from LDS ----
    float4 xv4 = ((const float4*)lds_x)[tid];
    float amax = fmaxf(fmaxf(fabsf(xv4.x), fabsf(xv4.y)),
                       fmaxf(fabsf(xv4.z), fabsf(xv4.w)));
#pragma unroll
    for (int off = 16; off > 0; off >>= 1)
        amax = fmaxf(amax, __shfl_xor(amax, off, 32));
    if (lane == 0) s_max[wid] = amax;
    __syncthreads();
    float rmax = s_max[0];
#pragma unroll
    for (int w = 1; w < NWAVE; ++w) rmax = fmaxf(rmax, s_max[w]);
    const float smax  = rmax / pcf;       // step_max = max|x| / positive_clip
    const float range = smax - STEP_MIN;

    // ---- Step-tile loop: each wave owns 16 candidate steps per tile ----
    unsigned long long best = ~0ull;      // (err_bits << 32) | idx

    const int m  = lane & 15;             // A-matrix row (step within tile)
    const int k0 = (lane >> 4) * 2;       // A-matrix K pair: lanes<16 -> K0,1; else K2,3
    const v2f bones = {1.0f, 1.0f};       // B == ones: layout-independent reduction

    for (int tile = wid; tile < N_TILES; tile += NWAVE) {
        int sidx = tile * 16 + m;
        int cidx = sidx < N_ITERS ? sidx : (N_ITERS - 1);   // tail lanes: dup last step
        float t    = (float)cidx / (float)(N_ITERS - 1);
        float step = STEP_MIN + range * t;
        float rstep = 1.0f / step;        // one IEEE div per candidate, hoisted

        v8f acc = {};
#pragma unroll 4
        for (int c0 = 0; c0 < C_DIM; c0 += 4) {
            float2 xp = *(const float2*)(&lds_x[c0 + k0]);  // 8B-aligned LDS read
            v2f a;
            a.x = sq_err(xp.x, step, rstep, ncf, pcf);
            a.y = sq_err(xp.y, step, rstep, ncf, pcf);
            // D[m,n] = sum_k A[m,k] * 1 + C[m,n]  -> 16 steps x 4 cols reduced/instr
            acc = __builtin_amdgcn_wmma_f32_16x16x4_f32(
                false, a, false, bones, (short)0, acc, false, false);
        }

        // Harvest: lane holds err for steps mbase..mbase+7 (replicated over N cols)
        const int mbase = tile * 16 + ((lane >> 4) << 3);
#pragma unroll
        for (int v = 0; v < 8; ++v) {
            int idx = mbase + v;
            unsigned long long key =
                ((unsigned long long)__float_as_uint(acc[v]) << 32) |
                (unsigned int)idx;
            key  = (idx < N_ITERS) ? key : ~0ull;   // mask tail steps
            best = (key < best) ? key : best;       // branchless umin
        }
    }

    // ---- Wave umin reduce, then block reduce ----
#pragma unroll
    for (int off = 16; off > 0; off >>= 1) {
        unsigned long long k2 = __shfl_xor(best, off, 32);
        best = (k2 < best) ? k2 : best;
    }
    if (lane == 0) s_key[wid] = best;
    __syncthreads();

    if (tid == 0) {
        unsigned long long b = s_key[0];
#pragma unroll
        for (int w = 1; w < NWAVE; ++w) {
            unsigned long long k2 = s_key[w];
            b = (k2 < b) ? k2 : b;
        }
        int bi = (int)(unsigned int)(b & 0xffffffffull);
        float t = (float)bi / (float)(N_ITERS - 1);
        out[row] = STEP_MIN + range * t;   // == steps[row, best]
    }
}

extern "C" void kernel_launch(void* const* d_in, const int* in_sizes, int n_in,
                              void* d_out, int out_size, void* d_ws, size_t ws_size,
                              hipStream_t stream) {
    const float* x   = (const float*)d_in[0];
    const int*   ncp = (const int*)d_in[1];   // negative_clip (-8)
    const int*   pcp = (const int*)d_in[2];   // positive_clip (7)
    float*       out = (float*)d_out;

    int R = in_sizes[0] / C_DIM;              // 1024 rows
    lsq_init_kernel<<<R, BLOCK, 0, stream>>>(x, ncp, pcp, out);
    (void)n_in; (void)out_size; (void)d_ws; (void)ws_size;
}